// StateScanPluginP0_64922725646539
// MI455X (gfx1250) — compile-verified
//
#include <hip/hip_runtime.h>

// Problem constants (from reference)
#define B_DIM   16
#define T_DIM   2048
#define NTOK    1024
#define DDIM    64
#define HDIM    256

#define CHUNK   128                 // tokens folded per phase-1 workgroup
#define NCHUNK  (T_DIM / CHUNK)     // 16 chunk-products per batch
#define LSTRIDE 68                  // LDS row stride (floats): 64+4 kills bank conflicts

typedef __attribute__((ext_vector_type(2))) float v2f;
typedef __attribute__((ext_vector_type(8))) float v8f;

// Issue one async global->LDS 128-bit copy (ASYNCcnt-tracked, no VGPR data).
__device__ __forceinline__ void async_copy_b128(uint32_t lds_byte_off,
                                                const void* gaddr)
{
    asm volatile("global_load_async_to_lds_b128 %0, %1, off"
                 :: "v"(lds_byte_off), "v"((uint64_t)(uintptr_t)gaddr)
                 : "memory");
}

// ---------------------------------------------------------------------------
// Ordered product of `count` 64x64 fp32 matrices:  P = M[count-1] @ ... @ M[0]
// ids != nullptr : matrix s is matBase + ids[s]*4096   (gather, phase 1)
// ids == nullptr : matrix s is matBase + s*4096        (contiguous, phase 2)
// Result (row-major 64x64) written to `out`.
// Block = 256 threads = 8 wave32. Wave w computes tile row ti=w/2, tile cols
// {2*(w&1), 2*(w&1)+1} per step via V_WMMA_F32_16X16X4_F32.
// ldsPT holds P transposed so the B fragment (P[k][n],P[k+1][n]) is one b64.
// A operand is double-buffered and filled by GLOBAL_LOAD_ASYNC_TO_LDS_B128
// issued one step ahead; s_wait_asynccnt 4 + barrier publishes it.
// ---------------------------------------------------------------------------
__device__ __forceinline__ void chain_product_device(
    const float* __restrict__ matBase,
    const int*   __restrict__ ids,
    int count,
    float* __restrict__ out)
{
    __shared__ float ldsA [2][DDIM * LSTRIDE];  // left operand, double-buffered
    __shared__ float ldsPT[DDIM * LSTRIDE];     // running product, TRANSPOSED

    const int tid  = threadIdx.x;
    const int lane = tid & 31;
    const int wave = tid >> 5;        // 0..7
    const int ti   = wave >> 1;       // C tile row   0..3
    const int tj0  = (wave & 1) * 2;  // C tile cols  tj0, tj0+1
    const int ln   = lane & 15;       // M/N index within tile
    const int lh   = lane >> 4;       // K-half select (WMMA f32 16x16x4 layout)

    const int mRow  = tid >> 2;        // cooperative loader: row 0..63
    const int cBase = (tid & 3) * 16;  // cooperative loader: col base

    // Per-thread LDS byte offsets of the two A staging slots (low 32 bits of
    // the generic pointer are the LDS offset on gfx1250's aperture scheme).
    const uint32_t aOff0 =
        (uint32_t)(uintptr_t)(void*)&ldsA[0][mRow * LSTRIDE + cBase];
    const uint32_t aOff1 =
        (uint32_t)(uintptr_t)(void*)&ldsA[1][mRow * LSTRIDE + cBase];

    // ---- prologue: kick async copy of A[1] into buffer 1 ----
    if (count > 1) {
        const float* src =
            matBase + (size_t)(ids ? ids[1] : 1) * (DDIM * DDIM);
        const float* g = src + mRow * DDIM + cBase;
        #pragma unroll
        for (int i = 0; i < 4; ++i)
            async_copy_b128(aOff1 + 16u * i, g + 4 * i);
    }

    // ---- first matrix -> ldsPT (transposed) ----
    {
        const float* src = matBase + (size_t)(ids ? ids[0] : 0) * (DDIM * DDIM);
        #pragma unroll
        for (int i = 0; i < 4; ++i) {
            float4 v = *(const float4*)(src + mRow * DDIM + cBase + 4 * i);
            int c = cBase + 4 * i;
            ldsPT[(c + 0) * LSTRIDE + mRow] = v.x;
            ldsPT[(c + 1) * LSTRIDE + mRow] = v.y;
            ldsPT[(c + 2) * LSTRIDE + mRow] = v.z;
            ldsPT[(c + 3) * LSTRIDE + mRow] = v.w;
        }
    }

    for (int s = 1; s < count; ++s) {
        const int buf = s & 1;

        // Issue async copy of A[s+1] into the other buffer (its last readers
        // finished before the barrier that ended step s-1), then wait for our
        // own step-s copies (4 ops in flight belong to step s+1).
        if (s + 1 < count) {
            const float* nsrc =
                matBase + (size_t)(ids ? ids[s + 1] : (s + 1)) * (DDIM * DDIM);
            const float* g = nsrc + mRow * DDIM + cBase;
            const uint32_t dst = buf ? aOff0 : aOff1;
            #pragma unroll
            for (int i = 0; i < 4; ++i)
                async_copy_b128(dst + 16u * i, g + 4 * i);

            if (s + 2 < count) {  // warm L2 two steps ahead (first HBM pass)
                const float* p =
                    matBase + (size_t)(ids ? ids[s + 2] : (s + 2)) * (DDIM * DDIM);
                __builtin_prefetch(p + mRow * DDIM + cBase, 0, 3);
            }
            asm volatile("s_wait_asynccnt 0x4" ::: "memory");
        } else {
            asm volatile("s_wait_asynccnt 0x0" ::: "memory");
        }

        __syncthreads();  // publishes: A[s] (all waves' async), PT from s-1

        // ---- C = A @ P : 16 K-steps of V_WMMA_F32_16X16X4_F32 per tile ----
        v8f acc0 = {};
        v8f acc1 = {};
        const float* aRow  = &ldsA[buf][(ti * 16 + ln) * LSTRIDE];
        const float* bRow0 = &ldsPT[( tj0      * 16 + ln) * LSTRIDE];
        const float* bRow1 = &ldsPT[((tj0 + 1) * 16 + ln) * LSTRIDE];
        #pragma unroll
        for (int kk = 0; kk < 16; ++kk) {
            const int k = kk * 4 + lh * 2;
            v2f a  = *(const v2f*)(aRow  + k);   // A[m][k..k+1] / A[m][k+2..k+3]
            v2f b0 = *(const v2f*)(bRow0 + k);   // (P[k][n], P[k+1][n]) via PT
            v2f b1 = *(const v2f*)(bRow1 + k);
            acc0 = __builtin_amdgcn_wmma_f32_16x16x4_f32(
                       false, a, false, b0, (short)0, acc0, false, false);
            acc1 = __builtin_amdgcn_wmma_f32_16x16x4_f32(
                       false, a, false, b1, (short)0, acc1, false, false);
        }

        __syncthreads();  // all P reads done

        // D tile layout: lane l, vgpr r -> C[ti*16 + lh*8 + r][tj*16 + ln]
        {
            float* p0 = &ldsPT[( tj0      * 16 + ln) * LSTRIDE + ti * 16 + lh * 8];
            float* p1 = &ldsPT[((tj0 + 1) * 16 + ln) * LSTRIDE + ti * 16 + lh * 8];
            *(float4*)(p0 + 0) = make_float4(acc0[0], acc0[1], acc0[2], acc0[3]);
            *(float4*)(p0 + 4) = make_float4(acc0[4], acc0[5], acc0[6], acc0[7]);
            *(float4*)(p1 + 0) = make_float4(acc1[0], acc1[1], acc1[2], acc1[3]);
            *(float4*)(p1 + 4) = make_float4(acc1[4], acc1[5], acc1[6], acc1[7]);
        }
    }

    __syncthreads();
    // un-transpose PT -> out (row-major)
    #pragma unroll
    for (int i = 0; i < 4; ++i) {
        int c = cBase + 4 * i;
        float4 v = make_float4(ldsPT[(c + 0) * LSTRIDE + mRow],
                               ldsPT[(c + 1) * LSTRIDE + mRow],
                               ldsPT[(c + 2) * LSTRIDE + mRow],
                               ldsPT[(c + 3) * LSTRIDE + mRow]);
        *(float4*)(out + mRow * DDIM + c) = v;
    }
}

// Phase 1: one WG per (batch, chunk) -> partial product of CHUNK matrices
__global__ void __launch_bounds__(256) chunk_products_kernel(
    const float* __restrict__ repr_M, const int* __restrict__ input_ids,
    float* __restrict__ partials)
{
    const int b = blockIdx.y;
    const int c = blockIdx.x;
    chain_product_device(repr_M,
                         input_ids + b * T_DIM + c * CHUNK,
                         CHUNK,
                         partials + (size_t)(b * NCHUNK + c) * (DDIM * DDIM));
}

// Phase 2: one WG per batch -> S[b] = partial[15] @ ... @ partial[0]
__global__ void __launch_bounds__(256) reduce_chunks_kernel(
    const float* __restrict__ partials, float* __restrict__ S)
{
    const int b = blockIdx.x;
    chain_product_device(partials + (size_t)b * NCHUNK * (DDIM * DDIM),
                         nullptr, NCHUNK,
                         S + (size_t)b * (DDIM * DDIM));
}

// Phase 3a: h = relu(S.flat @ W1 + b1)   (16 x 256)
__global__ void __launch_bounds__(256) mlp1_kernel(
    const float* __restrict__ S, const float* __restrict__ W1,
    const float* __restrict__ b1v, float* __restrict__ h)
{
    const int b = blockIdx.x;
    const int j = threadIdx.x;          // 0..255
    const float* s = S + (size_t)b * (DDIM * DDIM);
    float acc = b1v[j];
    #pragma unroll 4
    for (int k = 0; k < DDIM * DDIM; ++k)
        acc = fmaf(s[k], W1[(size_t)k * HDIM + j], acc);
    h[b * HDIM + j] = fmaxf(acc, 0.0f);
}

// Phase 3b: logits = h @ W2 + b2   (16 x 1024)
__global__ void __launch_bounds__(256) mlp2_kernel(
    const float* __restrict__ h, const float* __restrict__ W2,
    const float* __restrict__ b2v, float* __restrict__ out)
{
    const int b = blockIdx.y;
    const int i = blockIdx.x * 256 + threadIdx.x;   // 0..1023
    const float* hb = h + (size_t)b * HDIM;
    float acc = b2v[i];
    #pragma unroll 4
    for (int k = 0; k < HDIM; ++k)
        acc = fmaf(hb[k], W2[(size_t)k * NTOK + i], acc);
    out[b * NTOK + i] = acc;
}

extern "C" void kernel_launch(void* const* d_in, const int* in_sizes, int n_in,
                              void* d_out, int out_size, void* d_ws, size_t ws_size,
                              hipStream_t stream)
{
    const int*   input_ids = (const int*)  d_in[0];   // (16, 2048) int32
    const float* repr_M    = (const float*)d_in[1];   // (1024, 64, 64) f32
    const float* W1        = (const float*)d_in[2];   // (4096, 256)
    const float* b1        = (const float*)d_in[3];   // (256,)
    const float* W2        = (const float*)d_in[4];   // (256, 1024)
    const float* b2        = (const float*)d_in[5];   // (1024,)
    float* out = (float*)d_out;                       // (16, 1024) f32

    float* ws       = (float*)d_ws;
    float* partials = ws;                                               // 16*16*4096 f
    float* S        = partials + (size_t)B_DIM * NCHUNK * DDIM * DDIM;  // 16*4096 f
    float* h        = S + (size_t)B_DIM * DDIM * DDIM;                  // 16*256 f

    chunk_products_kernel<<<dim3(NCHUNK, B_DIM), 256, 0, stream>>>(
        repr_M, input_ids, partials);
    reduce_chunks_kernel<<<dim3(B_DIM), 256, 0, stream>>>(partials, S);
    mlp1_kernel<<<dim3(B_DIM), HDIM, 0, stream>>>(S, W1, b1, h);
    mlp2_kernel<<<dim3(NTOK / 256, B_DIM), 256, 0, stream>>>(h, W2, b2, out);
}